// GPTOSSAttentionQKSoftmaxBlock_27547920237056
// MI455X (gfx1250) — compile-verified
//
#include <hip/hip_runtime.h>
#include <math.h>

#define T_TOK    1024
#define HKV      8
#define MQ       8
#define NHEAD    (HKV * MQ)     // 64 flat heads
#define DHEAD    64
#define WINDOW   128
#define SM_SCALE 0.125f         // 1/sqrt(64)

typedef __attribute__((ext_vector_type(2))) float v2f;
typedef __attribute__((ext_vector_type(4))) float v4f;
typedef __attribute__((ext_vector_type(8))) float v8f;

// ---------------------------------------------------------------------------
// Pass 1: stream zeros over the whole 256 MB output (masked entries are
// exactly 0 after softmax).  Non-temporal b128 stores -> pure HBM stream,
// ~11 us at 23.3 TB/s; this is the kernel's roofline floor.
// ---------------------------------------------------------------------------
__global__ void zero_fill_kernel(float* __restrict__ out, long n4, long total) {
  v4f z = {0.f, 0.f, 0.f, 0.f};
  v4f* o = (v4f*)out;
  long idx    = (long)blockIdx.x * blockDim.x + threadIdx.x;
  long stride = (long)gridDim.x * blockDim.x;
  for (long i = idx; i < n4; i += stride)
    __builtin_nontemporal_store(z, o + i);
  if (idx == 0)                       // tail (none for this shape, but safe)
    for (long t = n4 * 4; t < total; ++t) out[t] = 0.f;
}

// ---------------------------------------------------------------------------
// Pass 2: one wave32 per (flat head, 16-query tile).  Band of 9 k-tiles
// (144 cols) covers [q0-128, q0+15] which contains every unmasked key for
// rows q0..q0+15.  Scores via exact-fp32 V_WMMA_F32_16X16X4_F32 chains.
//
// A (16x4 f32) fragment layout (ISA 7.12.2): lanes 0-15 hold K=0,1 in
// v0,v1; lanes 16-31 hold K=2,3.  B (4x16) mirrors it with N = lane&15.
// C/D (16x16): VGPR r -> M=r (lanes 0-15) / M=r+8 (lanes 16-31), N=lane&15.
// ---------------------------------------------------------------------------
__global__ __launch_bounds__(32)
void band_attn_kernel(const float* __restrict__ q,
                      const float* __restrict__ k,
                      const float* __restrict__ sinks,
                      float* __restrict__ out) {
  const int qt   = blockIdx.x;        // query tile 0..63
  const int head = blockIdx.y;        // flat head 0..63  (= hkv*MQ + m)
  const int hkv  = head >> 3;         // head / MQ
  const int lane = threadIdx.x;
  const int lo   = lane & 15;
  const int hi   = lane >> 4;         // 0 = low half, 1 = high half

  const int q0  = qt * 16;
  const int jlo = q0 - WINDOW;        // band start column (may be negative)

  __shared__ float sc[16][144];       // 9 KB

  // ---- Q A-fragments: 16 slices of K=4, invariant across k-tiles ----
  // q layout [T, HKV, M, D]: row stride = HKV*MQ*DHEAD, head offset = head*D
  const float* qrow = q + (long)(q0 + lo) * (HKV * MQ * DHEAD) + (long)head * DHEAD;
  v2f qa[16];
#pragma unroll
  for (int s = 0; s < 16; ++s) {
    const int d = s * 4 + hi * 2;     // contiguous pair -> global_load_b64
    qa[s] = *(const v2f*)(qrow + d);
  }

  // ---- accumulate 9 score tiles (16x16 each) ----
  v8f acc[9];
#pragma unroll
  for (int kt = 0; kt < 9; ++kt) {
    v8f c = {0.f, 0.f, 0.f, 0.f, 0.f, 0.f, 0.f, 0.f};
    const int k0 = jlo + kt * 16;     // wave-uniform -> scalar branch, EXEC stays full
    if (k0 >= 0) {
      // k layout [T, HKV, D]: B[kd][n] = K[k0+n][d], contiguous d pair
      const float* krow = k + (long)(k0 + lo) * (HKV * DHEAD) + (long)hkv * DHEAD;
#pragma unroll
      for (int s = 0; s < 16; ++s) {
        const int d = s * 4 + hi * 2;
        v2f b = *(const v2f*)(krow + d);
        c = __builtin_amdgcn_wmma_f32_16x16x4_f32(false, qa[s], false, b,
                                                  (short)0, c, false, false);
      }
    }
    acc[kt] = c;
  }

  // ---- scale + causal/window mask -> LDS ----
#pragma unroll
  for (int kt = 0; kt < 9; ++kt) {
    const int k0 = jlo + kt * 16;
    const int j  = k0 + lo;
#pragma unroll
    for (int r = 0; r < 8; ++r) {
      const int row = r + hi * 8;     // C/D layout: high half holds M=r+8
      const int i   = q0 + row;
      const float s = acc[kt][r] * SM_SCALE;
      const bool masked = (j < 0) | (j > i) | (j <= i - WINDOW);
      sc[row][kt * 16 + lo] = masked ? -__builtin_inff() : s;
    }
  }
  __syncthreads();

  // ---- softmax with sink logit; lanes (L, L+16) split row L&15 in half ----
  const int   row = lo;
  const int   i   = q0 + row;
  const int   c0  = hi * 72;
  const float snk = sinks[head];

  float mx = snk;
  for (int c = c0; c < c0 + 72; ++c) mx = fmaxf(mx, sc[row][c]);
  mx = fmaxf(mx, __shfl_xor(mx, 16, 32));

  float sum = 0.f;
  for (int c = c0; c < c0 + 72; ++c) {
    const float e = __expf(sc[row][c] - mx);   // exp(-inf)=0 handles mask
    sc[row][c] = e;
    sum += e;
  }
  sum += __shfl_xor(sum, 16, 32);
  const float inv = 1.0f / (sum + __expf(snk - mx));  // sink joins denominator only

  float* obase = out + (long)head * T_TOK * T_TOK + (long)i * T_TOK;
  for (int c = c0; c < c0 + 72; ++c) {
    const int j = jlo + c;
    if (j >= 0) obase[j] = sc[row][c] * inv;
  }
}

// ---------------------------------------------------------------------------
extern "C" void kernel_launch(void* const* d_in, const int* in_sizes, int n_in,
                              void* d_out, int out_size, void* d_ws, size_t ws_size,
                              hipStream_t stream) {
  const float* q     = (const float*)d_in[0];   // [T, HKV, M, D]
  const float* k     = (const float*)d_in[1];   // [T, HKV, D]
  const float* sinks = (const float*)d_in[2];   // [HKV*M]
  float*       out   = (float*)d_out;           // [HKV, M, T, T]

  const long total = (long)out_size;
  zero_fill_kernel<<<8192, 256, 0, stream>>>(out, total / 4, total);

  dim3 grid(T_TOK / 16, NHEAD);                 // 64 q-tiles x 64 heads
  band_attn_kernel<<<grid, 32, 0, stream>>>(q, k, sinks, out);
}